// GCN_35983236006362
// MI455X (gfx1250) — compile-verified
//
#include <hip/hip_runtime.h>

typedef __attribute__((ext_vector_type(2))) float v2f;
typedef __attribute__((ext_vector_type(8))) float v8f;

#define HDIM 64
#define LN_EPS 1e-5f
#define KP_STRIDE 80  // float2 stride per K-pair row: 160 dwords ≡ 32 mod 64 banks

__device__ __forceinline__ void atomAddF(float* p, float v) {
  __hip_atomic_fetch_add(p, v, __ATOMIC_RELAXED, __HIP_MEMORY_SCOPE_AGENT);
}

// ---------------------------------------------------------------- utilities
__global__ void fill_kernel(float* __restrict__ p, float v, int n) {
  int i = blockIdx.x * blockDim.x + threadIdx.x;
  if (i < n) p[i] = v;
}

__global__ void deg_count_kernel(const int* __restrict__ dst, float* __restrict__ deg, int ne) {
  int e = blockIdx.x * blockDim.x + threadIdx.x;
  if (e < ne) atomAddF(&deg[dst[e]], 1.0f);
}

__global__ void rsqrt_kernel(float* __restrict__ p, int n) {
  int i = blockIdx.x * blockDim.x + threadIdx.x;
  if (i < n) p[i] = rsqrtf(p[i]);
}

// ------------------------------------------------ conv1: [N,9] x [9,64] (tiny K, scalar)
__global__ void gemm_in9_kernel(const float* __restrict__ x, const float* __restrict__ W,
                                float* __restrict__ out, int nrows) {
  int idx = blockIdx.x * blockDim.x + threadIdx.x;
  if (idx >= nrows * HDIM) return;
  int node = idx >> 6, c = idx & 63;
  const float* xr = x + node * 9;
  float s = 0.f;
#pragma unroll
  for (int k = 0; k < 9; ++k) s = fmaf(xr[k], W[k * HDIM + c], s);
  out[idx] = s;
}

// ------------------------------------------------ edge scatter: agg[dst] += h[src]*norm
__global__ void scatter_kernel(const int* __restrict__ src, const int* __restrict__ dst,
                               const float* __restrict__ dis, const float* __restrict__ h,
                               float* __restrict__ agg, int ne) {
  long long t = (long long)blockIdx.x * blockDim.x + threadIdx.x;
  int e = (int)(t >> 4);
  if (e >= ne) return;
  int g = ((int)t & 15) * 4;
  int s = src[e], d = dst[e];
  float nrm = dis[s] * dis[d];
  const float4 hv = *(const float4*)(h + (size_t)s * HDIM + g);
  float* a = agg + (size_t)d * HDIM + g;
  atomAddF(a + 0, hv.x * nrm);
  atomAddF(a + 1, hv.y * nrm);
  atomAddF(a + 2, hv.z * nrm);
  atomAddF(a + 3, hv.w * nrm);
}

// ------------------------------------------------ self-loop term + bias (+relu)
template <bool RELU>
__global__ void combine_kernel(const float* __restrict__ h, const float* __restrict__ dis,
                               const float* __restrict__ bias, float* __restrict__ agg,
                               int nrows) {
  int idx = blockIdx.x * blockDim.x + threadIdx.x;
  if (idx >= nrows * HDIM) return;
  int node = idx >> 6, c = idx & 63;
  float di = dis[node];
  float v = agg[idx] + h[idx] * di * di + bias[c];
  if (RELU) v = fmaxf(v, 0.f);
  agg[idx] = v;
}

// ------------------------------------------------ WMMA f32 GEMM: [N,64] x [64,64]
// 4 waves/block, 16 rows/wave, 4x (16x16) f32 accumulators, K-step 4 via
// v_wmma_f32_16x16x4_f32. Weights staged in LDS as K-pairs (float2 per
// (k-pair, col)) so each B fragment is one aligned ds_load_b64, and the
// K-pair row stride (160 dwords) puts the two half-waves on disjoint banks.
template <bool BIAS, bool RELU, bool RES>
__global__ __launch_bounds__(128) void gemm64_kernel(
    const float* __restrict__ in, const float* __restrict__ W,
    const float* __restrict__ bias, const float* __restrict__ res,
    float* __restrict__ out, int nrows) {
  __shared__ v2f wl2[32 * KP_STRIDE];  // 20 KB
  for (int i = threadIdx.x; i < 32 * HDIM; i += 128) {
    int kp = i >> 6, col = i & 63;
    v2f p = {W[(2 * kp) * HDIM + col], W[(2 * kp + 1) * HDIM + col]};
    wl2[kp * KP_STRIDE + col] = p;
  }
  __syncthreads();

  const int lane = threadIdx.x & 31;
  const int wave = threadIdx.x >> 5;
  const int rowBase = blockIdx.x * 64 + wave * 16;
  const int m = lane & 15;   // row within tile (A) / col within tile (B,C)
  const int hi = lane >> 4;  // K-pair select
  int aRow = rowBase + m;
  if (aRow >= nrows) aRow = nrows - 1;  // clamp; tail stores predicated below
  const float* aPtr = in + (size_t)aRow * HDIM + hi * 2;

  v8f c0 = {}, c1 = {}, c2 = {}, c3 = {};
#pragma unroll
  for (int k0 = 0; k0 < HDIM; k0 += 4) {
    v2f a = *(const v2f*)(aPtr + k0);  // A: K=(k0,k0+1) lanes 0-15, (k0+2,k0+3) lanes 16-31
    const v2f* bp = &wl2[((k0 >> 1) + hi) * KP_STRIDE + m];
    v2f b0 = bp[0];
    v2f b1 = bp[16];
    v2f b2 = bp[32];
    v2f b3 = bp[48];
    c0 = __builtin_amdgcn_wmma_f32_16x16x4_f32(false, a, false, b0, (short)0, c0, false, false);
    c1 = __builtin_amdgcn_wmma_f32_16x16x4_f32(false, a, false, b1, (short)0, c1, false, false);
    c2 = __builtin_amdgcn_wmma_f32_16x16x4_f32(false, a, false, b2, (short)0, c2, false, false);
    c3 = __builtin_amdgcn_wmma_f32_16x16x4_f32(false, a, false, b3, (short)0, c3, false, false);
  }

  v8f acc[4] = {c0, c1, c2, c3};
#pragma unroll
  for (int t = 0; t < 4; ++t) {
    const int ccol = t * 16 + m;
    float bv = BIAS ? bias[ccol] : 0.f;
#pragma unroll
    for (int j = 0; j < 8; ++j) {
      int row = rowBase + hi * 8 + j;  // C/D layout: vgpr j -> M=j (lanes<16), M=8+j (lanes>=16)
      if (row < nrows) {
        float v = acc[t][j] + bv;
        if (RES) v += res[(size_t)row * HDIM + ccol];
        if (RELU) v = fmaxf(v, 0.f);
        out[(size_t)row * HDIM + ccol] = v;
      }
    }
  }
}

// ------------------------------------------------ global LayerNorm (graph mode)
__global__ void stats_kernel(const float* __restrict__ h, float* __restrict__ stats, int n) {
  __shared__ float ssum[256], ssq[256];
  float s = 0.f, q = 0.f;
  for (int i = blockIdx.x * blockDim.x + threadIdx.x; i < n; i += gridDim.x * blockDim.x) {
    float v = h[i];
    s += v;
    q += v * v;
  }
  ssum[threadIdx.x] = s;
  ssq[threadIdx.x] = q;
  __syncthreads();
  for (int off = 128; off > 0; off >>= 1) {
    if ((int)threadIdx.x < off) {
      ssum[threadIdx.x] += ssum[threadIdx.x + off];
      ssq[threadIdx.x] += ssq[threadIdx.x + off];
    }
    __syncthreads();
  }
  if (threadIdx.x == 0) {
    atomAddF(&stats[0], ssum[0]);
    atomAddF(&stats[1], ssq[0]);
  }
}

__global__ void applyln_kernel(float* __restrict__ h, const float* __restrict__ stats,
                               const float* __restrict__ w, const float* __restrict__ b,
                               int nrows) {
  int idx = blockIdx.x * blockDim.x + threadIdx.x;
  if (idx >= nrows * HDIM) return;
  float invN = 1.0f / ((float)nrows * (float)HDIM);
  float mu = stats[0] * invN;
  float var = stats[1] * invN - mu * mu;
  float r = rsqrtf(var + LN_EPS);
  int c = idx & 63;
  h[idx] = (h[idx] - mu) * r * w[c] + b[c];
}

// ------------------------------------------------ final projection + segment mean
__global__ void final_kernel(const float* __restrict__ h, const float* __restrict__ fw,
                             const float* __restrict__ fb, const int* __restrict__ batch,
                             float* __restrict__ gsum, float* __restrict__ gcnt, int nrows) {
  int n = blockIdx.x * blockDim.x + threadIdx.x;
  if (n >= nrows) return;
  const float4* hr = (const float4*)(h + (size_t)n * HDIM);
  float s = fb[0];
#pragma unroll
  for (int k = 0; k < 16; ++k) {
    float4 v = hr[k];
    float4 wv = ((const float4*)fw)[k];
    s += v.x * wv.x + v.y * wv.y + v.z * wv.z + v.w * wv.w;
  }
  int g = batch[n];
  atomAddF(&gsum[g], s);
  atomAddF(&gcnt[g], 1.f);
}

__global__ void div_kernel(const float* __restrict__ gsum, const float* __restrict__ gcnt,
                           float* __restrict__ out, int ng) {
  int g = blockIdx.x * blockDim.x + threadIdx.x;
  if (g < ng) out[g] = gsum[g] / fmaxf(gcnt[g], 1.0f);
}

// ---------------------------------------------------------------- launcher
extern "C" void kernel_launch(void* const* d_in, const int* in_sizes, int n_in,
                              void* d_out, int out_size, void* d_ws, size_t ws_size,
                              hipStream_t stream) {
  const float* x      = (const float*)d_in[0];
  const int*   edge   = (const int*)d_in[1];
  const int*   batch  = (const int*)d_in[2];
  const float* conv1W = (const float*)d_in[3];
  const float* conv1b = (const float*)d_in[4];
  const float* fc1W   = (const float*)d_in[5];
  const float* fc1b   = (const float*)d_in[6];
  const float* fc2W   = (const float*)d_in[7];
  const float* fc2b   = (const float*)d_in[8];
  const float* n1w    = (const float*)d_in[9];
  const float* n1b    = (const float*)d_in[10];
  const float* conv2W = (const float*)d_in[11];
  const float* conv2b = (const float*)d_in[12];
  const float* fc3W   = (const float*)d_in[13];
  const float* fc3b   = (const float*)d_in[14];
  const float* fc4W   = (const float*)d_in[15];
  const float* fc4b   = (const float*)d_in[16];
  const float* n2w    = (const float*)d_in[17];
  const float* n2b    = (const float*)d_in[18];
  const float* fcfW   = (const float*)d_in[19];
  const float* fcfb   = (const float*)d_in[20];

  const int N = in_sizes[2];       // nodes
  const int E = in_sizes[1] / 2;   // edges
  const int G = out_size;          // graphs
  const int* src  = edge;
  const int* dstp = edge + E;

  const size_t NH = (size_t)N * HDIM;
  float* P     = (float*)d_ws;
  float* Q     = P + NH;
  float* R     = Q + NH;
  float* dis   = R + NH;
  float* stats = dis + N;
  float* gsum  = stats + 2;
  float* gcnt  = gsum + G;

  const int TB = 256;
  auto nb = [](long long n, int tb) { return (unsigned)((n + tb - 1) / tb); };
  const int gb = (N + 63) / 64;  // GEMM blocks (64 rows/block)

  // degrees -> dis = rsqrt(deg + 1)
  fill_kernel<<<nb(N, TB), TB, 0, stream>>>(dis, 1.0f, N);
  deg_count_kernel<<<nb(E, TB), TB, 0, stream>>>(dstp, dis, E);
  rsqrt_kernel<<<nb(N, TB), TB, 0, stream>>>(dis, N);

  // conv1: P = x@W1 ; Q = relu(scatter(P) + P*dis^2 + b1)
  gemm_in9_kernel<<<nb((long long)N * HDIM, TB), TB, 0, stream>>>(x, conv1W, P, N);
  fill_kernel<<<nb((long long)NH, TB), TB, 0, stream>>>(Q, 0.f, (int)NH);
  scatter_kernel<<<nb((long long)E * 16, TB), TB, 0, stream>>>(src, dstp, dis, P, Q, E);
  combine_kernel<true><<<nb((long long)NH, TB), TB, 0, stream>>>(P, dis, conv1b, Q, N);

  // fc1 / fc2 (+residual Q)
  gemm64_kernel<true, true, false><<<gb, 128, 0, stream>>>(Q, fc1W, fc1b, nullptr, R, N);
  gemm64_kernel<true, true, true><<<gb, 128, 0, stream>>>(R, fc2W, fc2b, Q, P, N);

  // layernorm 1 (graph mode, global stats) in place on P
  fill_kernel<<<1, 2, 0, stream>>>(stats, 0.f, 2);
  stats_kernel<<<512, 256, 0, stream>>>(P, stats, (int)NH);
  applyln_kernel<<<nb((long long)NH, TB), TB, 0, stream>>>(P, stats, n1w, n1b, N);

  // conv2: Q = P@W2 ; R = scatter(Q) + Q*dis^2 + b2  (no relu)
  gemm64_kernel<false, false, false><<<gb, 128, 0, stream>>>(P, conv2W, nullptr, nullptr, Q, N);
  fill_kernel<<<nb((long long)NH, TB), TB, 0, stream>>>(R, 0.f, (int)NH);
  scatter_kernel<<<nb((long long)E * 16, TB), TB, 0, stream>>>(src, dstp, dis, Q, R, E);
  combine_kernel<false><<<nb((long long)NH, TB), TB, 0, stream>>>(Q, dis, conv2b, R, N);

  // fc3 / fc4 (+residual P)
  gemm64_kernel<true, true, false><<<gb, 128, 0, stream>>>(R, fc3W, fc3b, nullptr, Q, N);
  gemm64_kernel<true, false, true><<<gb, 128, 0, stream>>>(Q, fc4W, fc4b, P, R, N);

  // layernorm 2 in place on R
  fill_kernel<<<1, 2, 0, stream>>>(stats, 0.f, 2);
  stats_kernel<<<512, 256, 0, stream>>>(R, stats, (int)NH);
  applyln_kernel<<<nb((long long)NH, TB), TB, 0, stream>>>(R, stats, n2w, n2b, N);

  // final projection + per-graph mean
  fill_kernel<<<nb(G, TB), TB, 0, stream>>>(gsum, 0.f, G);
  fill_kernel<<<nb(G, TB), TB, 0, stream>>>(gcnt, 0.f, G);
  final_kernel<<<nb(N, TB), TB, 0, stream>>>(R, fcfW, fcfb, batch, gsum, gcnt, N);
  div_kernel<<<nb(G, TB), TB, 0, stream>>>(gsum, gcnt, (float*)d_out, G);
}